// TemporalAttDecoder_41094247088366
// MI455X (gfx1250) — compile-verified
//
#include <hip/hip_runtime.h>
#include <hip/hip_bf16.h>
#include <math.h>

// MI455X / gfx1250 (CDNA5, wave32, WGP, WMMA).
// Memory-bound problem: g = 256MB, read exactly twice (score pass + att pass)
// => ~22us floor at 23.3 TB/s. f32 WMMA (16x16x4) handles the 137 GFLOP of
// GEMM work without precision loss. 128-row A tile lives in LDS (CDNA5 gives
// 320KB/WGP); W1/Wp stream from L2 once per block (2GB total L2 traffic).

typedef __attribute__((ext_vector_type(2))) float v2f;
typedef __attribute__((ext_vector_type(8))) float v8f;

#define Bdim 512
#define Tdim 256
#define Gdim 512
#define Ydim 2
#define MROWS 128
#define LDA 513   // padded LDS row stride (floats) -> conflict-free A reads

// ---------------------------------------------------------------------------
// Kernel 1: fused  vi = tanh(g@W1)@w2 ,  vd = (tanh(g@Wp)) . gT[b]
//           score[b,t] = (vi + b2) * vd   (masked by t < n_day[b])
// grid = (B*T/MROWS) = 1024 blocks, 256 threads = 8 waves (wave32)
// ---------------------------------------------------------------------------
__global__ __launch_bounds__(256)
void tad_score_kernel(const float* __restrict__ g,
                      const float* __restrict__ gT,
                      const int*   __restrict__ n_day,
                      const float* __restrict__ W1,
                      const float* __restrict__ w2,
                      const float* __restrict__ b2,
                      const float* __restrict__ Wp,
                      float* __restrict__ score_ws) {
    extern __shared__ float lds[];
    float* Atile = lds;                    // MROWS * LDA
    float* viP   = Atile + MROWS * LDA;    // MROWS
    float* vdP   = viP + MROWS;            // MROWS
    float* w2s   = vdP + MROWS;            // Gdim
    float* gTs   = w2s + Gdim;             // Gdim

    const int tid    = threadIdx.x;
    const int wave   = tid >> 5;
    const int lane   = tid & 31;
    const int half   = lane >> 4;          // 0: lanes 0-15, 1: lanes 16-31
    const int lane16 = lane & 15;
    const int b      = blockIdx.x >> 1;
    const int t0     = (blockIdx.x & 1) * MROWS;

    const float* gBase = g + ((size_t)b * Tdim + t0) * Gdim;

    if (tid < MROWS) { viP[tid] = 0.0f; vdP[tid] = 0.0f; }
    for (int i = tid; i < Gdim; i += 256) {
        w2s[i] = w2[i];
        gTs[i] = gT[(size_t)b * Gdim + i];
    }
    // Load 128x512 f32 A tile into LDS (coalesced float4), padded rows.
    const float4* g4 = (const float4*)gBase;
    for (int i = tid; i < (MROWS * Gdim) / 4; i += 256) {
        const int row = i >> 7;            // 128 float4 per row
        const int k4  = (i & 127) << 2;
        float4 v = g4[i];
        float* dst = Atile + row * LDA + k4;
        dst[0] = v.x; dst[1] = v.y; dst[2] = v.z; dst[3] = v.w;
    }
    __syncthreads();

    const v8f vzero = {0.f, 0.f, 0.f, 0.f, 0.f, 0.f, 0.f, 0.f};

    for (int mat = 0; mat < 2; ++mat) {
        const float* W        = (mat == 0) ? W1 : Wp;
        const float* scaleVec = (mat == 0) ? w2s : gTs;
        float*       dstP     = (mat == 0) ? viP : vdP;

        for (int j = 0; j < 4; ++j) {           // wave owns 4 N-tiles (64 cols)
            const int nb   = (wave * 4 + j) * 16;
            const int ncol = nb + lane16;

            v8f acc[8];
            #pragma unroll
            for (int m = 0; m < 8; ++m) acc[m] = vzero;

            for (int kb = 0; kb < Gdim; kb += 32) {
                // B fragments for this K-chunk, cached in regs, reused by 8 M-tiles.
                // B(4x16) lane layout: K = r + 2*half, N = lane16.
                float Bf[16];
                #pragma unroll
                for (int w = 0; w < 8; ++w) {
                    #pragma unroll
                    for (int r = 0; r < 2; ++r)
                        Bf[2 * w + r] =
                            W[(size_t)(kb + 4 * w + r + 2 * half) * Gdim + ncol];
                }
                #pragma unroll
                for (int m = 0; m < 8; ++m) {
                    // A(16x4) lane layout: M = lane16, K = r + 2*half.
                    const float* Arow =
                        Atile + (m * 16 + lane16) * LDA + kb + 2 * half;
                    #pragma unroll
                    for (int w = 0; w < 8; ++w) {
                        v2f a = { Arow[4 * w], Arow[4 * w + 1] };
                        v2f bb = { Bf[2 * w], Bf[2 * w + 1] };
                        acc[m] = __builtin_amdgcn_wmma_f32_16x16x4_f32(
                            false, a, false, bb, (short)0, acc[m], false, false);
                    }
                }
            }
            // C layout: vgpr e, lane l -> M = e + 8*half, N = lane16.
            const float s = scaleVec[ncol];
            #pragma unroll
            for (int m = 0; m < 8; ++m) {
                #pragma unroll
                for (int e = 0; e < 8; ++e) {
                    float v = tanhf(acc[m][e]) * s;
                    // reduce across the 16 lanes sharing (half, e)
                    v += __shfl_xor(v, 1, 32);
                    v += __shfl_xor(v, 2, 32);
                    v += __shfl_xor(v, 4, 32);
                    v += __shfl_xor(v, 8, 32);
                    if (lane16 == 0)
                        atomicAdd(&dstP[m * 16 + e + 8 * half], v);  // ds_add_f32
                }
            }
        }
    }
    __syncthreads();

    if (tid < MROWS) {
        const int t = t0 + tid;
        float sc = (viP[tid] + b2[0]) * vdP[tid];
        if (t >= n_day[b]) sc = -1000000000.0f;
        score_ws[(size_t)b * Tdim + t] = sc;
    }
}

// ---------------------------------------------------------------------------
// Kernel 2: softmax over T -> v_stared ; att_c = sum_t v*g ; head softmax.
// grid = B blocks, 256 threads.
// ---------------------------------------------------------------------------
__global__ __launch_bounds__(256)
void tad_att_out_kernel(const float* __restrict__ g,
                        const float* __restrict__ gT,
                        const float* __restrict__ score_ws,
                        const float* __restrict__ Wo,
                        const float* __restrict__ bo,
                        float* __restrict__ out) {
    __shared__ float sm[256];
    __shared__ float vs[256];
    __shared__ float r0[256];
    __shared__ float r1[256];

    const int tid = threadIdx.x;
    const int b   = blockIdx.x;

    float s = score_ws[(size_t)b * Tdim + tid];
    sm[tid] = s;
    __syncthreads();
    for (int off = 128; off > 0; off >>= 1) {
        if (tid < off) sm[tid] = fmaxf(sm[tid], sm[tid + off]);
        __syncthreads();
    }
    const float mx = sm[0];
    __syncthreads();
    float e = expf(s - mx);
    sm[tid] = e;
    __syncthreads();
    for (int off = 128; off > 0; off >>= 1) {
        if (tid < off) sm[tid] += sm[tid + off];
        __syncthreads();
    }
    const float v = e / sm[0];
    vs[tid] = v;
    out[(size_t)Bdim * Ydim + (size_t)b * Tdim + tid] = v;   // v_stared
    __syncthreads();

    // att_c: thread handles columns tid and tid+256 (coalesced over g rows)
    const int c0 = tid, c1 = tid + 256;
    float a0 = 0.0f, a1 = 0.0f;
    const float* gb = g + (size_t)b * Tdim * Gdim;
    for (int t = 0; t < Tdim; ++t) {
        const float vt = vs[t];
        const float* gr = gb + (size_t)t * Gdim;
        a0 = fmaf(vt, gr[c0], a0);
        a1 = fmaf(vt, gr[c1], a1);
    }
    const float gt0 = gT[(size_t)b * Gdim + c0];
    const float gt1 = gT[(size_t)b * Gdim + c1];
    // feats = [att_c(512) | gT(512)] @ Wo(1024x2)
    float p0 = a0 * Wo[c0 * Ydim + 0] + a1 * Wo[c1 * Ydim + 0] +
               gt0 * Wo[(Gdim + c0) * Ydim + 0] + gt1 * Wo[(Gdim + c1) * Ydim + 0];
    float p1 = a0 * Wo[c0 * Ydim + 1] + a1 * Wo[c1 * Ydim + 1] +
               gt0 * Wo[(Gdim + c0) * Ydim + 1] + gt1 * Wo[(Gdim + c1) * Ydim + 1];
    r0[tid] = p0;
    r1[tid] = p1;
    __syncthreads();
    for (int off = 128; off > 0; off >>= 1) {
        if (tid < off) { r0[tid] += r0[tid + off]; r1[tid] += r1[tid + off]; }
        __syncthreads();
    }
    if (tid == 0) {
        const float z0 = r0[0] + bo[0];
        const float z1 = r1[0] + bo[1];
        const float m  = fmaxf(z0, z1);
        const float e0 = expf(z0 - m);
        const float e1 = expf(z1 - m);
        const float inv = 1.0f / (e0 + e1);
        out[(size_t)b * Ydim + 0] = e0 * inv;
        out[(size_t)b * Ydim + 1] = e1 * inv;
    }
}

extern "C" void kernel_launch(void* const* d_in, const int* in_sizes, int n_in,
                              void* d_out, int out_size, void* d_ws, size_t ws_size,
                              hipStream_t stream) {
    const float* g    = (const float*)d_in[0];
    // d_in[1] (y) is unused by the reference.
    const float* g_T  = (const float*)d_in[2];
    const int*   nday = (const int*)d_in[3];
    const float* W1   = (const float*)d_in[4];
    const float* w2   = (const float*)d_in[5];
    const float* b2   = (const float*)d_in[6];
    const float* Wp   = (const float*)d_in[7];
    const float* Wo   = (const float*)d_in[8];
    const float* bo   = (const float*)d_in[9];
    float* out   = (float*)d_out;
    float* score = (float*)d_ws;       // B*T floats of scratch

    const size_t lds1 = (size_t)(MROWS * LDA + 2 * MROWS + 2 * Gdim) * sizeof(float);
    // ~262KB dynamic LDS: legal on CDNA5 (320KB/WGP) but above the default cap.
    hipFuncSetAttribute((const void*)tad_score_kernel,
                        hipFuncAttributeMaxDynamicSharedMemorySize, (int)lds1);

    tad_score_kernel<<<dim3((Bdim * Tdim) / MROWS), dim3(256), lds1, stream>>>(
        g, g_T, nday, W1, w2, b2, Wp, score);

    tad_att_out_kernel<<<dim3(Bdim), dim3(256), 0, stream>>>(
        g, g_T, score, Wo, bo, out);
}